// T3A_70420283785639
// MI455X (gfx1250) — compile-verified
//
#include <hip/hip_runtime.h>

typedef __attribute__((ext_vector_type(2))) float v2f;
typedef __attribute__((ext_vector_type(8))) float v8f;

#define N_SUP   34500
#define DIM     512
#define NCLS    345
#define NCLS_PAD 352          // 22 tiles of 16 -> 11 groups of 32
#define B_ROWS  65536
#define EPSF    1e-12f

// ---------------------------------------------------------------------------
// Kernel 1: per-support inverse row norm: invn[i] = 1/max(||sup_i||, eps)
// ---------------------------------------------------------------------------
__global__ void __launch_bounds__(256) k_invnorm(const float* __restrict__ sup,
                                                 float* __restrict__ invn) {
    __shared__ float red[256];
    const int i = blockIdx.x;
    const int t = threadIdx.x;
    const float* row = sup + (size_t)i * DIM;
    float a = row[t];
    float b = row[t + 256];
    red[t] = a * a + b * b;
    __syncthreads();
    for (int s = 128; s > 0; s >>= 1) {
        if (t < s) red[t] += red[t + s];
        __syncthreads();
    }
    if (t == 0) invn[i] = 1.0f / fmaxf(sqrtf(red[0]), EPSF);
}

// ---------------------------------------------------------------------------
// Kernel 2: selection mask folded with invnorm.
// Stable-sort rank within class == 1 + #{ j : y[j]==y[i] && (ent[j],j) < (ent[i],i) }.
// scale[i] = selected ? invn[i] : 0
// ---------------------------------------------------------------------------
__global__ void __launch_bounds__(256) k_select(const float* __restrict__ ent,
                                                const int* __restrict__ yhat,
                                                const int* __restrict__ fK,
                                                const float* __restrict__ invn,
                                                float* __restrict__ scale) {
    const int i = blockIdx.x * blockDim.x + threadIdx.x;
    if (i >= N_SUP) return;
    const int   c = yhat[i];
    const float e = ent[i];
    const int   K = fK[0];
    int cnt = 0;
    for (int j = 0; j < N_SUP; ++j) {
        if (yhat[j] == c) {
            const float ej = ent[j];
            if (ej < e || (ej == e && j < i)) ++cnt;
        }
    }
    scale[i] = (cnt < K) ? invn[i] : 0.0f;
}

// ---------------------------------------------------------------------------
// Kernel 3: build column-normalized weights, stored TRANSPOSED:
//   Wt[c][d] = W[d][c] / max(||W[:,c]||, eps),  Wt is [NCLS_PAD][DIM]
// Pad classes (c >= 345) write zeros (workspace is poisoned).
// ---------------------------------------------------------------------------
__global__ void __launch_bounds__(256) k_weights(const float* __restrict__ sup,
                                                 const int* __restrict__ yhat,
                                                 const float* __restrict__ scale,
                                                 float* __restrict__ Wt) {
    __shared__ float red[256];
    const int c = blockIdx.x;
    const int t = threadIdx.x;
    float a0 = 0.0f, a1 = 0.0f;
    if (c < NCLS) {
        for (int i = 0; i < N_SUP; ++i) {
            if (yhat[i] == c) {                 // uniform across block
                const float s = scale[i];
                if (s != 0.0f) {
                    const float* row = sup + (size_t)i * DIM;
                    a0 += row[t] * s;
                    a1 += row[t + 256] * s;
                }
            }
        }
    }
    red[t] = a0 * a0 + a1 * a1;
    __syncthreads();
    for (int s = 128; s > 0; s >>= 1) {
        if (t < s) red[t] += red[t + s];
        __syncthreads();
    }
    const float inv = 1.0f / fmaxf(sqrtf(red[0]), EPSF);
    Wt[(size_t)c * DIM + t]       = a0 * inv;
    Wt[(size_t)c * DIM + t + 256] = a1 * inv;
}

// ---------------------------------------------------------------------------
// Kernel 4: out = z @ W via V_WMMA_F32_16X16X4_F32, 2x2 register blocking.
// One wave32 computes a 32x32 output region (4 accumulators); per K-step of 4:
// 2 A-fragment + 2 B-fragment b64 loads feed 4 independent WMMAs (1 load/WMMA).
// f32 A/B fragment layout (ISA 7.12.2): VGPR j, lane l -> M/N = l%16,
// K = j + 2*(l/16) => each lane loads a consecutive float2 at offset 2*(l/16).
// Epilogue: col0 = n0 + lm <= 335 < 345 always, so only the col1 (=col0+16)
// guard exists, hoisted to a single branch (one divergence point per wave).
// ---------------------------------------------------------------------------
__global__ void __launch_bounds__(256) k_gemm(const float* __restrict__ z,
                                              const float* __restrict__ Wt,
                                              float* __restrict__ out) {
    const int NG   = NCLS_PAD / 32;                       // 11 n-groups
    const int wave = (int)((blockIdx.x * blockDim.x + threadIdx.x) >> 5);
    const int lane = threadIdx.x & 31;
    const int m_g  = wave / NG;                           // 0..2047
    const int n_g  = wave % NG;                           // 0..10
    const int half = lane >> 4;       // 0: K pair {0,1}, 1: K pair {2,3}
    const int lm   = lane & 15;

    const int m0 = m_g * 32;
    const int n0 = n_g * 32;

    const float* zrow0 = z  + (size_t)(m0 + lm) * DIM + 2 * half;
    const float* zrow1 = zrow0 + (size_t)16 * DIM;
    const float* wrow0 = Wt + (size_t)(n0 + lm) * DIM + 2 * half;
    const float* wrow1 = wrow0 + (size_t)16 * DIM;

    v8f acc00 = {}, acc01 = {}, acc10 = {}, acc11 = {};
#pragma unroll 2
    for (int k = 0; k < DIM; k += 4) {
        v2f a0 = *(const v2f*)(zrow0 + k);
        v2f a1 = *(const v2f*)(zrow1 + k);
        v2f b0 = *(const v2f*)(wrow0 + k);
        v2f b1 = *(const v2f*)(wrow1 + k);
        acc00 = __builtin_amdgcn_wmma_f32_16x16x4_f32(false, a0, false, b0,
                                                      (short)0, acc00, false, false);
        acc01 = __builtin_amdgcn_wmma_f32_16x16x4_f32(false, a0, false, b1,
                                                      (short)0, acc01, false, false);
        acc10 = __builtin_amdgcn_wmma_f32_16x16x4_f32(false, a1, false, b0,
                                                      (short)0, acc10, false, false);
        acc11 = __builtin_amdgcn_wmma_f32_16x16x4_f32(false, a1, false, b1,
                                                      (short)0, acc11, false, false);
    }

    // D layout: VGPR r, lane l -> row = 8*(l/16) + r, col = l%16 (within tile)
    const int rbase = m0 + 8 * half;
    const int col0  = n0 + lm;        // always < NCLS (max 335)
    const int col1  = col0 + 16;      // may exceed NCLS only when n_g == 10

    float* o0 = out + (size_t)rbase * NCLS + col0;        // rows rbase..rbase+7
    float* o1 = out + (size_t)(rbase + 16) * NCLS + col0; // rows rbase+16..+23
#pragma unroll
    for (int r = 0; r < 8; ++r) {
        o0[(size_t)r * NCLS] = acc00[r];
        o1[(size_t)r * NCLS] = acc10[r];
    }
    if (col1 < NCLS) {
#pragma unroll
        for (int r = 0; r < 8; ++r) {
            o0[(size_t)r * NCLS + 16] = acc01[r];
            o1[(size_t)r * NCLS + 16] = acc11[r];
        }
    }
}

// ---------------------------------------------------------------------------
extern "C" void kernel_launch(void* const* d_in, const int* in_sizes, int n_in,
                              void* d_out, int out_size, void* d_ws, size_t ws_size,
                              hipStream_t stream) {
    const float* z    = (const float*)d_in[0];   // [65536, 512]
    const float* sup  = (const float*)d_in[1];   // [34500, 512]
    const float* ent  = (const float*)d_in[2];   // [34500]
    const int*   yhat = (const int*)d_in[3];     // [34500]
    const int*   fK   = (const int*)d_in[4];     // scalar
    float* out = (float*)d_out;                  // [65536, 345]

    // workspace layout: invn[N] | scale[N] | Wt[NCLS_PAD*DIM]   (~1 MB)
    float* invn  = (float*)d_ws;
    float* scale = invn + N_SUP;
    float* Wt    = scale + N_SUP;

    k_invnorm<<<N_SUP, 256, 0, stream>>>(sup, invn);
    k_select<<<(N_SUP + 255) / 256, 256, 0, stream>>>(ent, yhat, fK, invn, scale);
    k_weights<<<NCLS_PAD, 256, 0, stream>>>(sup, yhat, scale, Wt);

    const int waves  = (B_ROWS / 32) * (NCLS_PAD / 32);   // 2048 * 11 = 22528
    const int blocks = waves / 8;                          // 8 waves / 256-thr block
    k_gemm<<<blocks, 256, 0, stream>>>(z, Wt, out);
}